// GraphUNet_TopK_19524921328240
// MI455X (gfx1250) — compile-verified
//
#include <hip/hip_runtime.h>
#include <math.h>

typedef float v2f __attribute__((ext_vector_type(2)));
typedef float v8f __attribute__((ext_vector_type(8)));

// ---------------------------------------------------------------------------
// Tiled f32 GEMM using V_WMMA_F32_16X16X4_F32 + GLOBAL_LOAD_ASYNC_TO_LDS_B128
// with LDS double-buffering (async stage of chunk i+1 overlaps compute of i).
// C[M,N] = A[M,K] @ B[K,N] (+ bias[N]) (optional relu).
// Block: 256 threads = 8 waves. Block tile 64x128, K-chunk 32.
// Wave w owns a 2x2 grid of 16x16 C tiles:
//   row tiles { (w&1)*2, (w&1)*2+1 }, col tiles { (w>>1)*2, (w>>1)*2+1 }.
// Per kk-step: 2 A-frags + 2 B-frags feed 4 WMMAs (2 LDS dwords per WMMA).
// ---------------------------------------------------------------------------
#define TM 64
#define TN 128
#define TK 32
#define AST (TK + 4)   // 36 floats: 144B rows -> 16B aligned, conflict-free
#define BST (TN + 4)   // 132 floats: 528B rows -> 16B aligned

__device__ __forceinline__ void async_b128(unsigned lds_off, const float* g) {
    asm volatile("global_load_async_to_lds_b128 %0, %1, off"
                 :: "v"(lds_off), "v"(g) : "memory");
}
__device__ __forceinline__ void wait_async0() {
    asm volatile("s_wait_asynccnt 0x0" ::: "memory");
}

__global__ __launch_bounds__(256)
void k_gemm(const float* __restrict__ A, const float* __restrict__ B,
            float* __restrict__ C, int M, int N, int K,
            const float* __restrict__ bias, int relu_flag)
{
    __shared__ float As[2][TM][AST];
    __shared__ float Bs[2][TK][BST];

    const int tid  = threadIdx.x;
    const int lane = tid & 31;
    const int wave = tid >> 5;
    const int m0 = blockIdx.y * TM;
    const int n0 = blockIdx.x * TN;
    const int mtb = (wave & 1) * 2;      // row-tile base (0 or 2) of 4
    const int ntb = (wave >> 1) * 2;     // col-tile base (0,2,4,6) of 8
    const int half = lane >> 4;          // K split per ISA A/B layout
    const int l16  = lane & 15;

    // Async staging legal only for fully in-bounds, 16B-aligned tiles.
    const bool fullA = (m0 + TM <= M) && ((K & 3) == 0);
    const bool fullB = (n0 + TN <= N) && ((N & 3) == 0);

    // Stage one 64x32 A-chunk and 32x128 B-chunk into LDS buffer `buf`.
    auto stage = [&](int buf, int k0) {
        const bool fullK = (k0 + TK <= K);
        const unsigned ab = (unsigned)(size_t)(&As[buf][0][0]);
        const unsigned bb = (unsigned)(size_t)(&Bs[buf][0][0]);
        if (fullA && fullK) {
            #pragma unroll
            for (int q = 0; q < (TM * TK / 4) / 256; ++q) {   // 2 rounds
                int id  = q * 256 + tid;
                int r   = id >> 3, seg = id & 7;              // 8 x b128 per row
                async_b128(ab + (unsigned)(r * AST + seg * 4) * 4u,
                           A + (long)(m0 + r) * K + k0 + seg * 4);
            }
        } else {
            for (int i = tid; i < TM * TK; i += 256) {
                int r = i >> 5, c = i & 31;
                int gr = m0 + r, gc = k0 + c;
                As[buf][r][c] = (gr < M && gc < K) ? A[(long)gr * K + gc] : 0.0f;
            }
        }
        if (fullB && fullK) {
            #pragma unroll
            for (int q = 0; q < (TK * TN / 4) / 256; ++q) {   // 4 rounds
                int id  = q * 256 + tid;
                int r   = id >> 5, seg = id & 31;             // 32 x b128 per row
                async_b128(bb + (unsigned)(r * BST + seg * 4) * 4u,
                           B + (long)(k0 + r) * N + n0 + seg * 4);
            }
        } else {
            for (int i = tid; i < TK * TN; i += 256) {
                int r = i >> 7, c = i & 127;
                int gr = k0 + r, gc = n0 + c;
                Bs[buf][r][c] = (gr < K && gc < N) ? B[(long)gr * N + gc] : 0.0f;
            }
        }
    };

    v8f acc00 = {}, acc01 = {}, acc10 = {}, acc11 = {};

    // prologue: fill buffer 0
    stage(0, 0);
    wait_async0();
    __syncthreads();

    int buf = 0;
    for (int k0 = 0; k0 < K; k0 += TK, buf ^= 1) {
        // overlap: issue async stage of next chunk into the other buffer
        if (k0 + TK < K) stage(buf ^ 1, k0 + TK);
        // light L2 prefetch two chunks ahead
        if (k0 + 2 * TK < K) {
            __builtin_prefetch(A + (long)(m0 + (tid & 63)) * K + (k0 + 2 * TK), 0, 0);
            __builtin_prefetch(B + (long)(k0 + 2 * TK + (tid & 31)) * N + n0 + ((tid >> 5) << 4), 0, 0);
        }

        const int ar0 = mtb * 16 + l16;
        const int bc0 = ntb * 16 + l16;
        #pragma unroll
        for (int kk = 0; kk < TK; kk += 4) {
            const int ak = kk + 2 * half;   // lanes 0-15: K={kk,kk+1}; 16-31: {kk+2,kk+3}
            v2f a0 = { As[buf][ar0][ak],      As[buf][ar0][ak + 1] };
            v2f a1 = { As[buf][ar0 + 16][ak], As[buf][ar0 + 16][ak + 1] };
            v2f b0 = { Bs[buf][ak][bc0],      Bs[buf][ak + 1][bc0] };
            v2f b1 = { Bs[buf][ak][bc0 + 16], Bs[buf][ak + 1][bc0 + 16] };
            acc00 = __builtin_amdgcn_wmma_f32_16x16x4_f32(false, a0, false, b0,
                                                          (short)0, acc00, false, false);
            acc01 = __builtin_amdgcn_wmma_f32_16x16x4_f32(false, a0, false, b1,
                                                          (short)0, acc01, false, false);
            acc10 = __builtin_amdgcn_wmma_f32_16x16x4_f32(false, a1, false, b0,
                                                          (short)0, acc10, false, false);
            acc11 = __builtin_amdgcn_wmma_f32_16x16x4_f32(false, a1, false, b1,
                                                          (short)0, acc11, false, false);
        }
        // drain this wave's async writes (next buffer), then publish to block
        wait_async0();
        __syncthreads();
    }

    // C/D layout: VGPR r -> row r (lanes 0-15) / row r+8 (lanes 16-31)
    const int col0 = n0 + ntb * 16 + l16;
    const int col1 = col0 + 16;
    const float bia0 = (bias && col0 < N) ? bias[col0] : 0.0f;
    const float bia1 = (bias && col1 < N) ? bias[col1] : 0.0f;
    #pragma unroll
    for (int r = 0; r < 8; ++r) {
        int row0 = m0 + mtb * 16 + r + 8 * half;
        int row1 = row0 + 16;
        if (row0 < M) {
            if (col0 < N) {
                float v = acc00[r] + bia0;
                if (relu_flag) v = fmaxf(v, 0.0f);
                C[(long)row0 * N + col0] = v;
            }
            if (col1 < N) {
                float v = acc01[r] + bia1;
                if (relu_flag) v = fmaxf(v, 0.0f);
                C[(long)row0 * N + col1] = v;
            }
        }
        if (row1 < M) {
            if (col0 < N) {
                float v = acc10[r] + bia0;
                if (relu_flag) v = fmaxf(v, 0.0f);
                C[(long)row1 * N + col0] = v;
            }
            if (col1 < N) {
                float v = acc11[r] + bia1;
                if (relu_flag) v = fmaxf(v, 0.0f);
                C[(long)row1 * N + col1] = v;
            }
        }
    }
}

// ---------------------------------------------------------------------------
// Small helper kernels
// ---------------------------------------------------------------------------
__global__ void k_zero(float* p, long n) {
    long i = (long)blockIdx.x * blockDim.x + threadIdx.x;
    if (i < n) p[i] = 0.0f;
}

__global__ void k_scatter_edges(const int* __restrict__ ei, float* A, int E, int n) {
    int e = blockIdx.x * blockDim.x + threadIdx.x;
    if (e >= E) return;
    int src = ei[e];        // row 0 of edge_index
    int dst = ei[E + e];    // row 1
    atomicAdd(&A[(long)dst * n + src], 1.0f);
}

// deg[i] = sum_j A[i][j] + (A[i][i]==0 ? 2 : 0); dinv = deg>0 ? rsqrt(deg) : 0
__global__ __launch_bounds__(256)
void k_row_deg(const float* __restrict__ A, float* __restrict__ dinv, int n) {
    __shared__ float red[256];
    int row = blockIdx.x;
    const float* Ar = A + (long)row * n;
    float s = 0.0f;
    for (int j = threadIdx.x; j < n; j += 256) s += Ar[j];
    red[threadIdx.x] = s;
    __syncthreads();
    for (int o = 128; o > 0; o >>= 1) {
        if (threadIdx.x < o) red[threadIdx.x] += red[threadIdx.x + o];
        __syncthreads();
    }
    if (threadIdx.x == 0) {
        float deg = red[0] + (Ar[row] == 0.0f ? 2.0f : 0.0f);
        dinv[row] = deg > 0.0f ? rsqrtf(deg) : 0.0f;
    }
}

// An[i][j] = dinv[i] * (A[i][j] with improved-self-loop fix) * dinv[j]
__global__ void k_normalize(const float* __restrict__ A, const float* __restrict__ dinv,
                            float* __restrict__ An, int n) {
    long i = (long)blockIdx.x * blockDim.x + threadIdx.x;
    long total = (long)n * n;
    if (i >= total) return;
    int r = (int)(i / n), c = (int)(i % n);
    float v = A[i];
    if (r == c && v == 0.0f) v = 2.0f;
    An[i] = dinv[r] * v * dinv[c];
}

// D = A with diagonal forced to dval (augment input copy)
__global__ void k_copy_setdiag(const float* __restrict__ A, float* __restrict__ D,
                               int n, float dval) {
    long i = (long)blockIdx.x * blockDim.x + threadIdx.x;
    long total = (long)n * n;
    if (i >= total) return;
    int r = (int)(i / n), c = (int)(i % n);
    D[i] = (r == c) ? dval : A[i];
}

__global__ void k_set_diag(float* A, int n, float v) {
    int i = blockIdx.x * blockDim.x + threadIdx.x;
    if (i < n) A[(long)i * n + i] = v;
}

__global__ __launch_bounds__(256)
void k_wnorm(const float* __restrict__ w, int c, float* __restrict__ out) {
    __shared__ float red[256];
    float s = 0.0f;
    for (int i = threadIdx.x; i < c; i += 256) s += w[i] * w[i];
    red[threadIdx.x] = s;
    __syncthreads();
    for (int o = 128; o > 0; o >>= 1) {
        if (threadIdx.x < o) red[threadIdx.x] += red[threadIdx.x + o];
        __syncthreads();
    }
    if (threadIdx.x == 0) out[0] = sqrtf(red[0]);
}

__global__ __launch_bounds__(256)
void k_score(const float* __restrict__ x, const float* __restrict__ w,
             const float* __restrict__ wnorm, float* __restrict__ score,
             int n, int c) {
    __shared__ float red[256];
    int row = blockIdx.x;
    float s = 0.0f;
    for (int j = threadIdx.x; j < c; j += 256) s += x[(long)row * c + j] * w[j];
    red[threadIdx.x] = s;
    __syncthreads();
    for (int o = 128; o > 0; o >>= 1) {
        if (threadIdx.x < o) red[threadIdx.x] += red[threadIdx.x + o];
        __syncthreads();
    }
    if (threadIdx.x == 0) score[row] = tanhf(red[0] / wnorm[0]);
}

// Single-block bitonic sort, descending by score. n is a power of two <= 4096.
#define SORT_MAX 4096
__global__ __launch_bounds__(1024)
void k_bitonic_sort(const float* __restrict__ score, int n,
                    int* __restrict__ idx_out, float* __restrict__ val_out) {
    __shared__ float skey[SORT_MAX];
    __shared__ int   sidx[SORT_MAX];
    int tid = threadIdx.x;
    for (int i = tid; i < n; i += 1024) { skey[i] = score[i]; sidx[i] = i; }
    __syncthreads();
    for (int k2 = 2; k2 <= n; k2 <<= 1) {
        for (int j = k2 >> 1; j > 0; j >>= 1) {
            for (int i = tid; i < n; i += 1024) {
                int ixj = i ^ j;
                if (ixj > i) {
                    bool dir = ((i & k2) == 0);   // descending overall
                    float ki = skey[i], kj = skey[ixj];
                    bool sw = dir ? (ki < kj) : (ki > kj);
                    if (sw) {
                        skey[i] = kj; skey[ixj] = ki;
                        int t = sidx[i]; sidx[i] = sidx[ixj]; sidx[ixj] = t;
                    }
                }
            }
            __syncthreads();
        }
    }
    for (int i = tid; i < n; i += 1024) { idx_out[i] = sidx[i]; val_out[i] = skey[i]; }
}

// xg[i][j] = x[idx[i]][j] * vals[i]
__global__ void k_gather_x(const float* __restrict__ x, const int* __restrict__ idx,
                           const float* __restrict__ vals, float* __restrict__ xg,
                           int k, int c) {
    long i = (long)blockIdx.x * blockDim.x + threadIdx.x;
    long total = (long)k * c;
    if (i >= total) return;
    int r = (int)(i / c), j = (int)(i % c);
    xg[i] = x[(long)idx[r] * c + j] * vals[r];
}

// Ag[i][j] = A[idx[i]][idx[j]]
__global__ void k_gather_A(const float* __restrict__ A, const int* __restrict__ idx,
                           float* __restrict__ Ag, int k, int n) {
    long i = (long)blockIdx.x * blockDim.x + threadIdx.x;
    long total = (long)k * k;
    if (i >= total) return;
    int r = (int)(i / k), c = (int)(i % k);
    Ag[i] = A[(long)idx[r] * n + idx[c]];
}

// xc[n][2c]: left half = res, right half = 0
__global__ void k_concat_init(const float* __restrict__ res, float* __restrict__ xc,
                              int n, int c) {
    long i = (long)blockIdx.x * blockDim.x + threadIdx.x;
    long total = (long)n * 2 * c;
    if (i >= total) return;
    int r = (int)(i / (2 * c)), cc = (int)(i % (2 * c));
    xc[i] = (cc < c) ? res[(long)r * c + cc] : 0.0f;
}

// xc[perm[r]][c_res + j] = x[r][j],  j < c_res  (x rows have width c_x)
__global__ void k_unpool_scatter(const float* __restrict__ x, const int* __restrict__ perm,
                                 float* __restrict__ xc, int k, int c_res, int c_x) {
    long i = (long)blockIdx.x * blockDim.x + threadIdx.x;
    long total = (long)k * c_res;
    if (i >= total) return;
    int r = (int)(i / c_res), j = (int)(i % c_res);
    xc[(long)perm[r] * (2 * c_res) + c_res + j] = x[(long)r * c_x + j];
}

__global__ __launch_bounds__(64)
void k_softmax(const float* __restrict__ h, float* __restrict__ out, int n, int c) {
    __shared__ float red[64];
    int row = blockIdx.x, tid = threadIdx.x;
    const float* hr = h + (long)row * c;
    float m = -1e30f;
    for (int j = tid; j < c; j += 64) m = fmaxf(m, hr[j]);
    red[tid] = m; __syncthreads();
    for (int o = 32; o > 0; o >>= 1) {
        if (tid < o) red[tid] = fmaxf(red[tid], red[tid + o]);
        __syncthreads();
    }
    m = red[0]; __syncthreads();
    float s = 0.0f;
    for (int j = tid; j < c; j += 64) s += __expf(hr[j] - m);
    red[tid] = s; __syncthreads();
    for (int o = 32; o > 0; o >>= 1) {
        if (tid < o) red[tid] += red[tid + o];
        __syncthreads();
    }
    s = red[0];
    for (int j = tid; j < c; j += 64) out[(long)row * c + j] = __expf(hr[j] - m) / s;
}

// ---------------------------------------------------------------------------
// Host orchestration
// ---------------------------------------------------------------------------
static inline unsigned cdiv(long a, long b) { return (unsigned)((a + b - 1) / b); }

extern "C" void kernel_launch(void* const* d_in, const int* in_sizes, int n_in,
                              void* d_out, int out_size, void* d_ws, size_t ws_size,
                              hipStream_t stream) {
    (void)n_in; (void)out_size; (void)ws_size;
    const float* x_in = (const float*)d_in[0];
    const int*   ei   = (const int*)d_in[1];
    const int    E    = in_sizes[1] / 2;
    const float *Wd[5], *bd[5], *pw[4], *Wu[4], *bu[4];
    for (int i = 0; i < 5; ++i) { Wd[i] = (const float*)d_in[2 + 2 * i];
                                  bd[i] = (const float*)d_in[3 + 2 * i]; }
    for (int i = 0; i < 4; ++i)   pw[i] = (const float*)d_in[12 + i];
    for (int i = 0; i < 4; ++i) { Wu[i] = (const float*)d_in[16 + 2 * i];
                                  bu[i] = (const float*)d_in[17 + 2 * i]; }
    const float* Wo = (const float*)d_in[24];
    const float* bo = (const float*)d_in[25];
    float* out = (float*)d_out;

    const int N0 = 4096;
    const int ns[5] = {4096, 2048, 1024, 512, 256};
    const int cs[5] = {32, 64, 128, 256, 512};

    // ---- workspace layout (bump allocator over d_ws, float units) ----
    float* ws = (float*)d_ws;
    size_t off = 0;
    auto alloc = [&](size_t nf) { float* p = ws + off; off += nf; return p; };
    float* Asave[5]; for (int l = 0; l < 5; ++l) Asave[l] = alloc((size_t)ns[l] * ns[l]);
    float* xsave[5]; for (int l = 0; l < 5; ++l) xsave[l] = alloc((size_t)ns[l] * cs[l]);
    float* An   = alloc((size_t)N0 * N0);   // normalized adjacency OR augment input copy
    float* Aaug = alloc((size_t)N0 * N0);   // A' @ A' result
    float* xw   = alloc((size_t)N0 * 512);  // x @ W intermediate
    float* xg   = alloc((size_t)N0 * 512);  // pooled / gathered x
    float* xc   = alloc((size_t)N0 * 512);  // concat buffer (up path)
    float* xb0  = alloc((size_t)N0 * 512);  // ping
    float* xb1  = alloc((size_t)N0 * 512);  // pong
    float* dinv = alloc(N0);
    float* score = alloc(N0);
    float* wn    = alloc(16);
    float* vals[4]; int* perms[4];
    for (int i = 0; i < 4; ++i) vals[i]  = alloc(N0);
    for (int i = 0; i < 4; ++i) perms[i] = (int*)alloc(N0);

    // gcn: outX[n,cout] = norm(A) @ (X[n,cin] @ W[cin,cout]) + bias (opt relu)
    auto gcn = [&](const float* A, int n, const float* X, int cin,
                   const float* W, int cout, const float* bias, int relu,
                   float* outX) {
        k_row_deg<<<n, 256, 0, stream>>>(A, dinv, n);
        k_normalize<<<cdiv((long)n * n, 256), 256, 0, stream>>>(A, dinv, An, n);
        dim3 g1(cdiv(cout, TN), cdiv(n, TM));
        k_gemm<<<g1, 256, 0, stream>>>(X, W, xw, n, cout, cin, nullptr, 0);
        dim3 g2(cdiv(cout, TN), cdiv(n, TM));
        k_gemm<<<g2, 256, 0, stream>>>(An, xw, outX, n, cout, n, bias, relu);
    };

    // ---- build A0 from edge list ----
    k_zero<<<cdiv((long)N0 * N0, 256), 256, 0, stream>>>(Asave[0], (long)N0 * N0);
    k_scatter_edges<<<cdiv(E, 256), 256, 0, stream>>>(ei, Asave[0], E, N0);

    // ---- level 0 GCN ----
    gcn(Asave[0], N0, x_in, 128, Wd[0], cs[0], bd[0], 1, xsave[0]);

    // ---- down path: levels 1..4 ----
    for (int l = 1; l <= 4; ++l) {
        int np = ns[l - 1], k = ns[l], cp = cs[l - 1];
        // augment: A' = A w/ diag=1 ; Aaug = A'@A' ; diag(Aaug)=0
        k_copy_setdiag<<<cdiv((long)np * np, 256), 256, 0, stream>>>(Asave[l - 1], An, np, 1.0f);
        dim3 ga(cdiv(np, TN), cdiv(np, TM));
        k_gemm<<<ga, 256, 0, stream>>>(An, An, Aaug, np, np, np, nullptr, 0);
        k_set_diag<<<cdiv(np, 256), 256, 0, stream>>>(Aaug, np, 0.0f);
        // top-k pooling (k = np/2, np power of two)
        k_wnorm<<<1, 256, 0, stream>>>(pw[l - 1], cp, wn);
        k_score<<<np, 256, 0, stream>>>(xsave[l - 1], pw[l - 1], wn, score, np, cp);
        k_bitonic_sort<<<1, 1024, 0, stream>>>(score, np, perms[l - 1], vals[l - 1]);
        k_gather_x<<<cdiv((long)k * cp, 256), 256, 0, stream>>>(
            xsave[l - 1], perms[l - 1], vals[l - 1], xg, k, cp);
        k_gather_A<<<cdiv((long)k * k, 256), 256, 0, stream>>>(
            Aaug, perms[l - 1], Asave[l], k, np);
        // level-l GCN
        gcn(Asave[l], k, xg, cp, Wd[l], cs[l], bd[l], 1, xsave[l]);
    }

    // ---- up path ----
    const float* cur = xsave[4];
    int ccur = cs[4];
    for (int i = 0; i < 4; ++i) {
        int j = 3 - i;
        int n = ns[j], cres = cs[j], kk = ns[j + 1];
        k_concat_init<<<cdiv((long)n * 2 * cres, 256), 256, 0, stream>>>(xsave[j], xc, n, cres);
        k_unpool_scatter<<<cdiv((long)kk * cres, 256), 256, 0, stream>>>(
            cur, perms[j], xc, kk, cres, ccur);
        float* outx = (i & 1) ? xb1 : xb0;
        gcn(Asave[j], n, xc, 2 * cres, Wu[i], cres, bu[i], (i < 3) ? 1 : 0, outx);
        cur = outx;
        ccur = cres;
    }

    // ---- final GCN (A = As[0]) + softmax ----
    k_row_deg<<<N0, 256, 0, stream>>>(Asave[0], dinv, N0);
    k_normalize<<<cdiv((long)N0 * N0, 256), 256, 0, stream>>>(Asave[0], dinv, An, N0);
    {
        dim3 g1(cdiv(37, TN), cdiv(N0, TM));
        k_gemm<<<g1, 256, 0, stream>>>(cur, Wo, xw, N0, 37, 32, nullptr, 0);
        dim3 g2(cdiv(37, TN), cdiv(N0, TM));
        k_gemm<<<g2, 256, 0, stream>>>(An, xw, xb0, N0, 37, N0, bo, 0);
    }
    k_softmax<<<N0, 64, 0, stream>>>(xb0, out, N0, 37);
}